// CARAFE_2250562863598
// MI455X (gfx1250) — compile-verified
//
#include <hip/hip_runtime.h>
#include <hip/hip_bf16.h>

// CARAFE for MI455X (gfx1250, wave32).
//   K1: 1x1 conv (GEMM 64x256x4608) via V_WMMA_F32_16X16X4_F32 + BN + SiLU
//   K2: 3x3 conv (implicit GEMM 112x576x4608) via WMMA + BN
//   K3: per-pixel softmax(25) + reassembly (gather/FMA, L2-resident)

typedef __attribute__((ext_vector_type(2))) float v2f;
typedef __attribute__((ext_vector_type(8))) float v8f;

#define N_BATCH 2
#define C_IN    256
#define C_MID   64
#define C_ENC   100
#define HW      48
#define P_SZ    (HW * HW)          // 2304 pixels per image
#define NP      (N_BATCH * P_SZ)   // 4608 GEMM columns
#define EPS     1e-5f

// ---------------------------------------------------------------------------
// Kernel 1: t = SiLU(BN1(w_comp @ X))   [M=64, K=256, N=4608]
// blockDim = 128 (4 waves); wave w computes M-tile rows [16w,16w+16).
// One block per 16-column N tile (288 blocks). 2304 % 16 == 0 so a column
// tile never crosses the batch boundary.
// ---------------------------------------------------------------------------
__global__ void __launch_bounds__(128)
carafe_comp_gemm(const float* __restrict__ X, const float* __restrict__ Wc,
                 const float* __restrict__ g1, const float* __restrict__ b1,
                 const float* __restrict__ m1, const float* __restrict__ v1,
                 float* __restrict__ T)
{
    const int lane = threadIdx.x & 31;
    const int wave = threadIdx.x >> 5;
    const int m0   = wave * 16;
    const int col0 = blockIdx.x * 16;
    const int n    = col0 / P_SZ;
    const int p0   = col0 % P_SZ;
    const int rowL = lane & 15;     // A row / C column lane
    const int kh   = lane >> 4;     // K-half select (0: K+0/K+1, 1: K+2/K+3)

    const float* Ab = Wc + (size_t)(m0 + rowL) * C_IN;           // A[row][k]
    const float* Bb = X + (size_t)n * (C_IN * P_SZ) + p0 + rowL; // B[k][col]

    v8f acc = {};
    for (int k = 0; k < C_IN; k += 4) {
        const int ka = k + 2 * kh;
        v2f a, b;
        a.x = Ab[ka];
        a.y = Ab[ka + 1];
        b.x = Bb[(size_t)ka * P_SZ];
        b.y = Bb[(size_t)(ka + 1) * P_SZ];
        acc = __builtin_amdgcn_wmma_f32_16x16x4_f32(
            false, a, false, b, (short)0, acc, false, false);
    }

    // C layout: VGPR r -> row (r + 8*kh), col = lane&15. Fuse BN + SiLU.
    #pragma unroll
    for (int r = 0; r < 8; ++r) {
        const int co = m0 + r + 8 * kh;
        const float sc = g1[co] * rsqrtf(v1[co] + EPS);
        const float sh = b1[co] - m1[co] * sc;
        float v = acc[r] * sc + sh;
        v = v * (1.0f / (1.0f + __expf(-v)));                    // SiLU
        T[(size_t)n * (C_MID * P_SZ) + (size_t)co * P_SZ + p0 + rowL] = v;
    }
}

// ---------------------------------------------------------------------------
// Kernel 2: e = BN2(w_enc *conv3x3* t)  as implicit GEMM
//   M=112 (100 padded), K=576 (=64*9), N=4608.  w_enc (ce,ci,ky,kx) layout
//   makes A[row][k] = w_enc[row*576 + k] exactly.
// grid = (288 N-tiles, 7 M-tiles), one wave per block.
// ---------------------------------------------------------------------------
__device__ __forceinline__ float enc_b_elem(const float* __restrict__ Tb,
                                            int kk, int y, int xcol)
{
    const int ci  = kk / 9;
    const int r9  = kk - ci * 9;
    const int dy  = r9 / 3 - 1;
    const int dx  = r9 - (r9 / 3) * 3 - 1;
    const int yy  = y + dy;
    const int xx  = xcol + dx;
    const bool ok = (yy >= 0) & (yy < HW) & (xx >= 0) & (xx < HW);
    return ok ? Tb[(size_t)ci * P_SZ + yy * HW + xx] : 0.0f;
}

__global__ void __launch_bounds__(32)
carafe_enc_gemm(const float* __restrict__ T, const float* __restrict__ We,
                const float* __restrict__ g2, const float* __restrict__ b2,
                const float* __restrict__ m2, const float* __restrict__ v2,
                float* __restrict__ E)
{
    const int lane = threadIdx.x & 31;
    const int m0   = blockIdx.y * 16;
    const int col0 = blockIdx.x * 16;
    const int n    = col0 / P_SZ;
    const int p0   = col0 % P_SZ;
    const int y    = p0 / HW;
    const int x0   = p0 % HW;
    const int rowL = lane & 15;
    const int kh   = lane >> 4;
    const int xcol = x0 + rowL;

    const int mrowA     = m0 + rowL;
    const bool a_valid  = (mrowA < C_ENC);
    const float* Ab     = We + (size_t)(a_valid ? mrowA : 0) * 576;
    const float* Tb     = T + (size_t)n * (C_MID * P_SZ);

    v8f acc = {};
    for (int k = 0; k < 576; k += 4) {
        const int ka = k + 2 * kh;
        v2f a, b;
        a.x = a_valid ? Ab[ka]     : 0.0f;
        a.y = a_valid ? Ab[ka + 1] : 0.0f;
        b.x = enc_b_elem(Tb, ka,     y, xcol);
        b.y = enc_b_elem(Tb, ka + 1, y, xcol);
        acc = __builtin_amdgcn_wmma_f32_16x16x4_f32(
            false, a, false, b, (short)0, acc, false, false);
    }

    #pragma unroll
    for (int r = 0; r < 8; ++r) {
        const int ce = m0 + r + 8 * kh;
        if (ce < C_ENC) {
            const float sc = g2[ce] * rsqrtf(v2[ce] + EPS);
            const float sh = b2[ce] - m2[ce] * sc;
            E[(size_t)n * (C_ENC * P_SZ) + (size_t)ce * P_SZ + p0 + rowL] =
                acc[r] * sc + sh;
        }
    }
}

// ---------------------------------------------------------------------------
// Kernel 3: softmax over 25 taps + reassembly.
// Algebra: out[n,c,2y+sy,2x+sx] = sum_{i,j} Wk[sub][5i+j] * X[n,c,y+i-2,x+j-2]
// (taps are independent of sy/sx), where e-channel for tap k, sub = 4k+sub.
// One block per source pixel (n,y,x); 256 threads = 256 channels; the 4
// softmaxes are computed once in LDS and shared.
// ---------------------------------------------------------------------------
__global__ void __launch_bounds__(256)
carafe_apply(const float* __restrict__ X, const float* __restrict__ E,
             float* __restrict__ Out)
{
    __shared__ float ev[4][25];
    __shared__ float wk[4][25];

    const int pix = blockIdx.x;            // 0..4607
    const int n   = pix / P_SZ;
    const int p   = pix % P_SZ;
    const int y   = p / HW;
    const int x   = p % HW;
    const int tid = threadIdx.x;

    if (tid < C_ENC) {                     // ce = 4*k + sub
        ev[tid & 3][tid >> 2] = E[(size_t)n * (C_ENC * P_SZ) + (size_t)tid * P_SZ + p];
    }
    __syncthreads();

    if (tid < 4) {
        float mx = ev[tid][0];
        #pragma unroll
        for (int k = 1; k < 25; ++k) mx = fmaxf(mx, ev[tid][k]);
        float sum = 0.0f;
        float e_[25];
        #pragma unroll
        for (int k = 0; k < 25; ++k) { e_[k] = __expf(ev[tid][k] - mx); sum += e_[k]; }
        const float inv = 1.0f / sum;
        #pragma unroll
        for (int k = 0; k < 25; ++k) wk[tid][k] = e_[k] * inv;
    }
    __syncthreads();

    const int c = tid;                     // one channel per thread
    const float* Xc = X + ((size_t)(n * C_IN + c)) * P_SZ;
    float o0 = 0.0f, o1 = 0.0f, o2 = 0.0f, o3 = 0.0f;

    #pragma unroll
    for (int i = 0; i < 5; ++i) {
        const int yy = y + i - 2;
        const bool vy = (yy >= 0) & (yy < HW);
        #pragma unroll
        for (int j = 0; j < 5; ++j) {
            const int xx = x + j - 2;
            const bool ok = vy & (xx >= 0) & (xx < HW);
            const float val = ok ? Xc[yy * HW + xx] : 0.0f;
            const int kk = i * 5 + j;
            o0 = fmaf(wk[0][kk], val, o0);
            o1 = fmaf(wk[1][kk], val, o1);
            o2 = fmaf(wk[2][kk], val, o2);
            o3 = fmaf(wk[3][kk], val, o3);
        }
    }

    const size_t base = (((size_t)(n * C_IN + c)) * 96 + 2 * y) * 96 + 2 * x;
    Out[base]      = o0;   // (sy,sx)=(0,0)
    Out[base + 1]  = o1;   // (0,1)
    Out[base + 96] = o2;   // (1,0)
    Out[base + 97] = o3;   // (1,1)
}

// ---------------------------------------------------------------------------
extern "C" void kernel_launch(void* const* d_in, const int* in_sizes, int n_in,
                              void* d_out, int out_size, void* d_ws, size_t ws_size,
                              hipStream_t stream) {
    const float* X      = (const float*)d_in[0];
    const float* w_comp = (const float*)d_in[1];
    const float* g1     = (const float*)d_in[2];
    const float* b1     = (const float*)d_in[3];
    const float* m1     = (const float*)d_in[4];
    const float* v1     = (const float*)d_in[5];
    const float* w_enc  = (const float*)d_in[6];
    const float* g2     = (const float*)d_in[7];
    const float* b2     = (const float*)d_in[8];
    const float* m2     = (const float*)d_in[9];
    const float* v2     = (const float*)d_in[10];
    float* Out = (float*)d_out;

    // Workspace layout: T (2*64*2304 f32 = 1.18 MB) | E (2*100*2304 f32 = 1.84 MB)
    float* T = (float*)d_ws;
    float* E = T + (size_t)N_BATCH * C_MID * P_SZ;

    carafe_comp_gemm<<<NP / 16, 128, 0, stream>>>(X, w_comp, g1, b1, m1, v1, T);
    carafe_enc_gemm<<<dim3(NP / 16, 7), 32, 0, stream>>>(T, w_enc, g2, b2, m2, v2, E);
    carafe_apply<<<NP, 256, 0, stream>>>(X, E, Out);
}